// Ultra_5781025980890
// MI455X (gfx1250) — compile-verified
//
#include <hip/hip_runtime.h>
#include <hip/hip_bf16.h>

// ---------------------------------------------------------------------------
// Ultra (NBFNet-style) relational GNN for MI455X / gfx1250.
// f32 end-to-end; GEMMs via V_WMMA_F32_16X16X4_F32; scatter via native
// no-return GLOBAL_ATOMIC_ADD_F32 (agent scope, relaxed).
// ---------------------------------------------------------------------------

typedef __attribute__((ext_vector_type(2))) float v2f;
typedef __attribute__((ext_vector_type(8))) float v8f;

#define EPSN 1e-5f

static __device__ __forceinline__ v8f wmma4(v2f a, v2f b, v8f c) {
    // D = A(16x4,f32) * B(4x16,f32) + C(16x16,f32)
    return __builtin_amdgcn_wmma_f32_16x16x4_f32(false, a, false, b, (short)0, c,
                                                 false, false);
}

// Native HW f32 atomic add, no return value -> GLOBAL_ATOMIC_ADD_F32
// (no-return form, tracked on STOREcnt), device/agent scope, relaxed order.
static __device__ __forceinline__ void atomic_add_f32(float* p, float v) {
    (void)__hip_atomic_fetch_add(p, v, __ATOMIC_RELAXED, __HIP_MEMORY_SCOPE_AGENT);
}

// --------------------------- utility kernels -------------------------------

__global__ void ultra_zero_kernel(float4* __restrict__ p, int n4) {
    int i = blockIdx.x * blockDim.x + threadIdx.x;
    if (i < n4) p[i] = make_float4(0.f, 0.f, 0.f, 0.f);
}

// boundary for relation graph: ones(D) at row q_rel[b].  1 block x 256 thr.
__global__ void ultra_set_query_ones_kernel(float* __restrict__ buf,
                                            const int* __restrict__ batch,
                                            int nNodes) {
    int b = threadIdx.x >> 6;   // 0..3
    int c = threadIdx.x & 63;
    int q = batch[b * 768 + 2]; // r_index = batch[b,0,2]
    buf[((size_t)b * nNodes + q) * 64 + c] = 1.0f;
}

// boundary for entity graph: rel_repr[b, r_index[b]] at row h_index[b].
__global__ void ultra_set_query_ent_kernel(float* __restrict__ buf,
                                           const float* __restrict__ rel_repr,
                                           const int* __restrict__ batch,
                                           int nNodes) {
    int b = threadIdx.x >> 6;
    int c = threadIdx.x & 63;
    int h = batch[b * 768 + 0];
    int r = batch[b * 768 + 2];
    buf[((size_t)b * nNodes + h) * 64 + c] = rel_repr[(r + b * 200) * 64 + c];
}

// --------------------------- edge scatter ----------------------------------
// One wave32 per (batch, edge); 2 channels per lane (float2 loads),
// two native global_atomic_add_f32 (no return) per lane.
__global__ void __launch_bounds__(256)
ultra_scatter_kernel(const float* __restrict__ x, const float* __restrict__ rel,
                     const int* __restrict__ ei, const int* __restrict__ et,
                     float* __restrict__ agg, int nE, int nNodes, int nB,
                     int relBstride) {
    int wid  = (blockIdx.x * blockDim.x + threadIdx.x) >> 5;
    int lane = threadIdx.x & 31;
    if (wid >= nB * nE) return;
    int b = wid / nE;
    int e = wid - b * nE;
    int src = ei[e];
    int dst = ei[nE + e];
    int ty  = et[e];
    const float2 xv = *(const float2*)(x + ((size_t)b * nNodes + src) * 64 + lane * 2);
    const float2 rv = *(const float2*)(rel + (size_t)b * relBstride + ty * 64 + lane * 2);
    float* a = agg + ((size_t)b * nNodes + dst) * 64 + lane * 2;
    atomic_add_f32(a + 0, xv.x * rv.x);
    atomic_add_f32(a + 1, xv.y * rv.y);
}

// ------------------- fused conv GEMM + LN + ReLU + residual ----------------
// out_row = relu(LN(concat(x_row, agg_row) @ W + bias; g, bt)) + x_row
// One wave per 16-row strip; W (128x64) staged in LDS; fp32 WMMA 16x16x4.
__global__ void __launch_bounds__(256)
ultra_conv_kernel(const float* __restrict__ x, const float* __restrict__ agg,
                  const float* __restrict__ W, const float* __restrict__ bias,
                  const float* __restrict__ g, const float* __restrict__ bt,
                  float* __restrict__ xout, int nrows) {
    __shared__ float Wlds[128 * 64];
    {
        const float4* s = (const float4*)W;
        float4* d = (float4*)Wlds;
        #pragma unroll
        for (int i = 0; i < (128 * 64 / 4) / 256; ++i)
            d[threadIdx.x + i * 256] = s[threadIdx.x + i * 256];
    }
    __syncthreads();

    int wave = threadIdx.x >> 5;
    int lane = threadIdx.x & 31;
    int strip = blockIdx.x * 8 + wave;
    int row0 = strip * 16;
    if (row0 >= nrows) return;

    int half = lane >> 4;   // A holds K=k+2*half..+1; C half selects rows 0-7/8-15
    int l15  = lane & 15;

    v8f acc[4];
    #pragma unroll
    for (int t = 0; t < 4; ++t)
        #pragma unroll
        for (int r = 0; r < 8; ++r) acc[t][r] = 0.f;

    const float* arow_x = x   + (size_t)(row0 + l15) * 64;
    const float* arow_g = agg + (size_t)(row0 + l15) * 64;

    // K = 0..63 from x
    #pragma unroll 2
    for (int k = 0; k < 64; k += 4) {
        const int kk = k + 2 * half;
        v2f a;
        { float2 t2 = *(const float2*)(arow_x + kk); a[0] = t2.x; a[1] = t2.y; }
        #pragma unroll
        for (int t = 0; t < 4; ++t) {
            v2f bf;
            bf[0] = Wlds[kk * 64 + t * 16 + l15];
            bf[1] = Wlds[(kk + 1) * 64 + t * 16 + l15];
            acc[t] = wmma4(a, bf, acc[t]);
        }
    }
    // K = 64..127 from agg
    #pragma unroll 2
    for (int k = 0; k < 64; k += 4) {
        const int kk = k + 2 * half;
        v2f a;
        { float2 t2 = *(const float2*)(arow_g + kk); a[0] = t2.x; a[1] = t2.y; }
        #pragma unroll
        for (int t = 0; t < 4; ++t) {
            v2f bf;
            bf[0] = Wlds[(64 + kk) * 64 + t * 16 + l15];
            bf[1] = Wlds[(64 + kk + 1) * 64 + t * 16 + l15];
            acc[t] = wmma4(a, bf, acc[t]);
        }
    }

    float bv[4], gv[4], btv[4];
    #pragma unroll
    for (int t = 0; t < 4; ++t) {
        int c = t * 16 + l15;
        bv[t] = bias[c]; gv[t] = g[c]; btv[t] = bt[c];
    }
    #pragma unroll
    for (int t = 0; t < 4; ++t)
        #pragma unroll
        for (int r = 0; r < 8; ++r) acc[t][r] += bv[t];

    // LayerNorm per output row: row m spread over 16-lane half x 4 fragments.
    #pragma unroll
    for (int r = 0; r < 8; ++r) {
        float s = acc[0][r] + acc[1][r] + acc[2][r] + acc[3][r];
        float q = acc[0][r] * acc[0][r] + acc[1][r] * acc[1][r] +
                  acc[2][r] * acc[2][r] + acc[3][r] * acc[3][r];
        #pragma unroll
        for (int m = 1; m <= 8; m <<= 1) {   // stays within the 16-lane half
            s += __shfl_xor(s, m, 32);
            q += __shfl_xor(q, m, 32);
        }
        float mean = s * (1.0f / 64.0f);
        float var  = q * (1.0f / 64.0f) - mean * mean;
        float inv  = rsqrtf(var + EPSN);
        int row = row0 + r + half * 8;
        #pragma unroll
        for (int t = 0; t < 4; ++t) {
            int c = t * 16 + l15;
            float v = (acc[t][r] - mean) * inv * gv[t] + btv[t];
            v = fmaxf(v, 0.f);
            xout[(size_t)row * 64 + c] = v + x[(size_t)row * 64 + c];
        }
    }
}

// --------------------- generic K=64 WMMA GEMM (+bias, opt relu) ------------
__global__ void __launch_bounds__(256)
ultra_gemm64_kernel(const float* __restrict__ in, const float* __restrict__ W,
                    const float* __restrict__ bias, float* __restrict__ out,
                    int nrows, int do_relu) {
    __shared__ float Wlds[64 * 64];
    {
        const float4* s = (const float4*)W;
        float4* d = (float4*)Wlds;
        #pragma unroll
        for (int i = 0; i < (64 * 64 / 4) / 256; ++i)
            d[threadIdx.x + i * 256] = s[threadIdx.x + i * 256];
    }
    __syncthreads();

    int wave = threadIdx.x >> 5;
    int lane = threadIdx.x & 31;
    int strip = blockIdx.x * 8 + wave;
    int row0 = strip * 16;
    if (row0 >= nrows) return;
    int half = lane >> 4;
    int l15  = lane & 15;

    v8f acc[4];
    #pragma unroll
    for (int t = 0; t < 4; ++t)
        #pragma unroll
        for (int r = 0; r < 8; ++r) acc[t][r] = 0.f;

    const float* arow = in + (size_t)(row0 + l15) * 64;
    #pragma unroll 2
    for (int k = 0; k < 64; k += 4) {
        const int kk = k + 2 * half;
        v2f a;
        { float2 t2 = *(const float2*)(arow + kk); a[0] = t2.x; a[1] = t2.y; }
        #pragma unroll
        for (int t = 0; t < 4; ++t) {
            v2f bf;
            bf[0] = Wlds[kk * 64 + t * 16 + l15];
            bf[1] = Wlds[(kk + 1) * 64 + t * 16 + l15];
            acc[t] = wmma4(a, bf, acc[t]);
        }
    }

    #pragma unroll
    for (int r = 0; r < 8; ++r) {
        int row = row0 + r + half * 8;
        #pragma unroll
        for (int t = 0; t < 4; ++t) {
            int c = t * 16 + l15;
            float v = acc[t][r] + bias[c];
            if (do_relu) v = fmaxf(v, 0.f);
            out[(size_t)row * 64 + c] = v;
        }
    }
}

// ------------------------------ final MLP ----------------------------------
// One block (128 thr) per (b,k): feat=[x[b,t], query[b]] (128);
// h = relu(feat@W1 + b1); out = h@W2 + b2.
__global__ void __launch_bounds__(128)
ultra_score_kernel(const float* __restrict__ x, const float* __restrict__ rel_repr,
                   const int* __restrict__ batch, const float* __restrict__ W1,
                   const float* __restrict__ b1, const float* __restrict__ W2,
                   const float* __restrict__ b2, float* __restrict__ out,
                   int nNodes) {
    int bk = blockIdx.x;
    int b = bk >> 8;
    int k = bk & 255;
    int t = batch[b * 768 + k * 3 + 1];
    int r = batch[b * 768 + 2];
    __shared__ float feat[128];
    __shared__ float red[4];
    int j = threadIdx.x;
    if (j < 64) feat[j] = x[((size_t)b * nNodes + t) * 64 + j];
    else        feat[j] = rel_repr[(r + b * 200) * 64 + (j - 64)];
    __syncthreads();
    float a = b1[j];
    #pragma unroll 4
    for (int i = 0; i < 128; ++i) a = fmaf(feat[i], W1[i * 128 + j], a);
    a = fmaxf(a, 0.f) * W2[j];
    #pragma unroll
    for (int m = 1; m <= 16; m <<= 1) a += __shfl_xor(a, m, 32);
    if ((j & 31) == 0) red[j >> 5] = a;
    __syncthreads();
    if (j == 0) out[bk] = red[0] + red[1] + red[2] + red[3] + b2[0];
}

// ------------------------------- launcher ----------------------------------
extern "C" void kernel_launch(void* const* d_in, const int* in_sizes, int n_in,
                              void* d_out, int out_size, void* d_ws, size_t ws_size,
                              hipStream_t stream) {
    const int B = 4, K = 256, N = 20000, E = 200000, Rn = 200, Er = 4000, L = 6;

    const int* batch   = (const int*)d_in[0];
    const int* ei      = (const int*)d_in[1];
    const int* et      = (const int*)d_in[2];
    const int* rel_ei  = (const int*)d_in[3];
    const int* rel_et  = (const int*)d_in[4];
    const float* rel_emb = (const float*)d_in[7];
    const float* Wr  = (const float*)d_in[8];
    const float* br  = (const float*)d_in[9];
    const float* gr  = (const float*)d_in[10];
    const float* btr = (const float*)d_in[11];
    const float* Wp1 = (const float*)d_in[12];
    const float* bp1 = (const float*)d_in[13];
    const float* Wp2 = (const float*)d_in[14];
    const float* bp2 = (const float*)d_in[15];
    const float* We  = (const float*)d_in[16];
    const float* be  = (const float*)d_in[17];
    const float* ge  = (const float*)d_in[18];
    const float* bte = (const float*)d_in[19];
    const float* W1  = (const float*)d_in[20];
    const float* b1  = (const float*)d_in[21];
    const float* W2  = (const float*)d_in[22];
    const float* b2  = (const float*)d_in[23];

    // workspace layout (floats)
    const size_t nRel = (size_t)B * Rn * 64;   //   51 200
    const size_t nEnt = (size_t)B * N * 64;    // 5 120 000
    const size_t need = (4 * nRel + 2 * nEnt) * sizeof(float);
    if (ws_size < need) return;  // insufficient scratch: safe no-op
    float* x_rel   = (float*)d_ws;
    float* agg_rel = x_rel + nRel;
    float* tmp_p   = agg_rel + nRel;
    float* rel_l   = tmp_p + nRel;
    float* x_ent   = rel_l + nRel;
    float* agg_ent = x_ent + nEnt;

    auto cdiv = [](int a, int b) { return (a + b - 1) / b; };
    const int relN4 = (int)(nRel / 4), entN4 = (int)(nEnt / 4);

    // ---------------- relation-graph model ----------------
    ultra_zero_kernel<<<cdiv(relN4, 256), 256, 0, stream>>>((float4*)x_rel, relN4);
    ultra_set_query_ones_kernel<<<1, 256, 0, stream>>>(x_rel, batch, Rn);
    const int relStrips = cdiv((B * Rn) / 16, 8);      // 800 rows -> 50 strips
    for (int l = 0; l < L; ++l) {
        ultra_zero_kernel<<<cdiv(relN4, 256), 256, 0, stream>>>((float4*)agg_rel, relN4);
        ultra_set_query_ones_kernel<<<1, 256, 0, stream>>>(agg_rel, batch, Rn);
        ultra_scatter_kernel<<<cdiv(B * Er, 8), 256, 0, stream>>>(
            x_rel, rel_emb, rel_ei, rel_et, agg_rel, Er, Rn, B, /*relBstride=*/0);
        ultra_conv_kernel<<<relStrips, 256, 0, stream>>>(
            x_rel, agg_rel, Wr + (size_t)l * 128 * 64, br + l * 64, gr + l * 64,
            btr + l * 64, x_rel, B * Rn);
    }

    // ---------------- entity-graph model ----------------
    ultra_zero_kernel<<<cdiv(entN4, 256), 256, 0, stream>>>((float4*)x_ent, entN4);
    ultra_set_query_ent_kernel<<<1, 256, 0, stream>>>(x_ent, x_rel, batch, N);
    const int entStrips = cdiv((B * N) / 16, 8);       // 80 000 rows -> 625 blocks
    for (int l = 0; l < L; ++l) {
        // per-layer relation projection: relu(rel_repr@Wp1+bp1)@Wp2+bp2
        ultra_gemm64_kernel<<<relStrips, 256, 0, stream>>>(
            x_rel, Wp1 + (size_t)l * 64 * 64, bp1 + l * 64, tmp_p, B * Rn, 1);
        ultra_gemm64_kernel<<<relStrips, 256, 0, stream>>>(
            tmp_p, Wp2 + (size_t)l * 64 * 64, bp2 + l * 64, rel_l, B * Rn, 0);
        ultra_zero_kernel<<<cdiv(entN4, 256), 256, 0, stream>>>((float4*)agg_ent, entN4);
        ultra_set_query_ent_kernel<<<1, 256, 0, stream>>>(agg_ent, x_rel, batch, N);
        ultra_scatter_kernel<<<cdiv(B * E, 8), 256, 0, stream>>>(
            x_ent, rel_l, ei, et, agg_ent, E, N, B, /*relBstride=*/Rn * 64);
        ultra_conv_kernel<<<entStrips, 256, 0, stream>>>(
            x_ent, agg_ent, We + (size_t)l * 128 * 64, be + l * 64, ge + l * 64,
            bte + l * 64, x_ent, B * N);
    }

    // ---------------- scoring MLP ----------------
    ultra_score_kernel<<<B * K, 128, 0, stream>>>(
        x_ent, x_rel, batch, W1, b1, W2, b2, (float*)d_out, N);
}